// EdgeMLP_78030965834143
// MI455X (gfx1250) — compile-verified
//
#include <hip/hip_runtime.h>
#include <hip/hip_bf16.h>

typedef float v2f __attribute__((ext_vector_type(2)));
typedef float v8f __attribute__((ext_vector_type(8)));

#define LDS_PITCH 132          // floats per W1 row in LDS (64-bank conflict-free for b64 reads)
#define LDS_PITCH4 (LDS_PITCH / 4)

// ---------------------------------------------------------------------------
// Phase 1: A = x @ W1[:, :64]^T ; B = x @ W1[:, 64:]^T   (per-node, WMMA fp32)
// One wave computes a 16-node x 128-output tile via V_WMMA_F32_16X16X4_F32.
// ---------------------------------------------------------------------------
__global__ void __launch_bounds__(256)
edgemlp_precompute_wmma(const float* __restrict__ x,
                        const float* __restrict__ W1,
                        float* __restrict__ A,
                        float* __restrict__ B,
                        int ntiles) {
    __shared__ __align__(16) float w1s[64 * LDS_PITCH];

    // Stage W1 [64][128] into LDS (padded pitch 132 floats).
    const float4* W14 = (const float4*)W1;
    for (int i = threadIdx.x; i < 64 * 32; i += blockDim.x) {
        int row = i >> 5;
        int c4  = i & 31;
        ((float4*)w1s)[row * LDS_PITCH4 + c4] = W14[row * 32 + c4];
    }
    __syncthreads();

    const int wave = threadIdx.x >> 5;
    const int lane = threadIdx.x & 31;
    const int tile = blockIdx.x * (blockDim.x >> 5) + wave;
    if (tile >= ntiles) return;           // wave-uniform: EXEC stays all-ones for WMMA

    const int node0 = tile * 16;
    const int mrow  = lane & 15;          // M index (A) / N index (B,C,D)
    const int khalf = lane >> 4;          // which K pair this half-wave holds

    const float* xrow = x + (size_t)(node0 + mrow) * 64;

    v8f accA[4] = {v8f{}, v8f{}, v8f{}, v8f{}};
    v8f accB[4] = {v8f{}, v8f{}, v8f{}, v8f{}};

    #pragma unroll
    for (int kk = 0; kk < 16; ++kk) {
        const int k0 = kk * 4 + khalf * 2;        // this lane's 2 K-values
        // A fragment: lane holds x[node0+mrow][k0], x[node0+mrow][k0+1]
        v2f afrag = *(const v2f*)(xrow + k0);
        #pragma unroll
        for (int j = 0; j < 4; ++j) {
            // B fragment: lane holds W1[j*16+mrow][cb + k0 .. +1]  (B[k][n], n=lane%16)
            const float* wrow = &w1s[(j * 16 + mrow) * LDS_PITCH];
            v2f bfragA = *(const v2f*)(wrow + k0);        // columns 0..63  -> A output
            v2f bfragB = *(const v2f*)(wrow + 64 + k0);   // columns 64..127-> B output
            accA[j] = __builtin_amdgcn_wmma_f32_16x16x4_f32(
                false, afrag, false, bfragA, (short)0, accA[j], false, false);
            accB[j] = __builtin_amdgcn_wmma_f32_16x16x4_f32(
                false, afrag, false, bfragB, (short)0, accB[j], false, false);
        }
    }

    // C/D layout: VGPR r, lanes0-15 -> M=r, lanes16-31 -> M=r+8; N = lane%16.
    #pragma unroll
    for (int j = 0; j < 4; ++j) {
        #pragma unroll
        for (int r = 0; r < 8; ++r) {
            const size_t row = (size_t)(node0 + r + 8 * khalf);
            A[row * 64 + j * 16 + mrow] = accA[j][r];
            B[row * 64 + j * 16 + mrow] = accB[j][r];
        }
    }
}

// Scalar tail for N % 16 leftover nodes (one thread per (node, out-feature)).
__global__ void edgemlp_precompute_tail(const float* __restrict__ x,
                                        const float* __restrict__ W1,
                                        float* __restrict__ A,
                                        float* __restrict__ B,
                                        int nstart, int nrem) {
    int t = blockIdx.x * blockDim.x + threadIdx.x;
    if (t >= nrem * 64) return;
    int node = nstart + (t >> 6);
    int j    = t & 63;
    const float* xr = x + (size_t)node * 64;
    const float* w  = W1 + (size_t)j * 128;
    float sa = 0.f, sb = 0.f;
    #pragma unroll 8
    for (int k = 0; k < 64; ++k) {
        sa = fmaf(xr[k], w[k], sa);
        sb = fmaf(xr[k], w[64 + k], sb);
    }
    A[(size_t)node * 64 + j] = sa;
    B[(size_t)node * 64 + j] = sb;
}

// ---------------------------------------------------------------------------
// Phase 2: out[e] = relu(A[src] + B[dst] + b1) . w2 + b2   (L2-resident gathers)
// ---------------------------------------------------------------------------
__global__ void __launch_bounds__(256)
edgemlp_edge_kernel(const long long* __restrict__ ei,
                    const float* __restrict__ A,
                    const float* __restrict__ B,
                    const float* __restrict__ b1,
                    const float* __restrict__ W2,
                    const float* __restrict__ b2,
                    float* __restrict__ out,
                    int E) {
    __shared__ float sb1[64];
    __shared__ float sw2[64];
    if (threadIdx.x < 64)        sb1[threadIdx.x]      = b1[threadIdx.x];
    else if (threadIdx.x < 128)  sw2[threadIdx.x - 64] = W2[threadIdx.x - 64];
    __syncthreads();

    const int e = blockIdx.x * blockDim.x + threadIdx.x;
    if (e >= E) return;

    const int s = (int)ei[e];
    const int d = (int)ei[(size_t)E + e];

    const float4* a4 = (const float4*)(A + (size_t)s * 64);
    const float4* g4 = (const float4*)(B + (size_t)d * 64);
    const float4* c4 = (const float4*)sb1;
    const float4* w4 = (const float4*)sw2;

    float acc = 0.f;
    #pragma unroll 4
    for (int j = 0; j < 16; ++j) {
        float4 av = a4[j];
        float4 bv = g4[j];
        float4 cv = c4[j];
        float4 wv = w4[j];
        acc = fmaf(fmaxf(av.x + bv.x + cv.x, 0.f), wv.x, acc);
        acc = fmaf(fmaxf(av.y + bv.y + cv.y, 0.f), wv.y, acc);
        acc = fmaf(fmaxf(av.z + bv.z + cv.z, 0.f), wv.z, acc);
        acc = fmaf(fmaxf(av.w + bv.w + cv.w, 0.f), wv.w, acc);
    }
    out[e] = acc + b2[0];
}

// Fallback if the workspace is too small for A/B: fully fused per-edge MLP.
__global__ void edgemlp_fused_fallback(const float* __restrict__ x,
                                       const long long* __restrict__ ei,
                                       const float* __restrict__ W1,
                                       const float* __restrict__ b1,
                                       const float* __restrict__ W2,
                                       const float* __restrict__ b2,
                                       float* __restrict__ out,
                                       int E) {
    const int e = blockIdx.x * blockDim.x + threadIdx.x;
    if (e >= E) return;
    const int s = (int)ei[e];
    const int d = (int)ei[(size_t)E + e];
    const float* xs = x + (size_t)s * 64;
    const float* xd = x + (size_t)d * 64;
    float acc = 0.f;
    for (int j = 0; j < 64; ++j) {
        const float* w = W1 + (size_t)j * 128;
        float h = b1[j];
        #pragma unroll 8
        for (int k = 0; k < 64; ++k) {
            h = fmaf(xs[k], w[k], h);
            h = fmaf(xd[k], w[64 + k], h);
        }
        acc = fmaf(fmaxf(h, 0.f), W2[j], acc);
    }
    out[e] = acc + b2[0];
}

extern "C" void kernel_launch(void* const* d_in, const int* in_sizes, int n_in,
                              void* d_out, int out_size, void* d_ws, size_t ws_size,
                              hipStream_t stream) {
    const float*     x  = (const float*)d_in[0];
    const long long* ei = (const long long*)d_in[1];   // int64 edge_index [2, E]
    const float*     W1 = (const float*)d_in[2];       // [64, 128]
    const float*     b1 = (const float*)d_in[3];       // [64]
    const float*     W2 = (const float*)d_in[4];       // [1, 64]
    const float*     b2 = (const float*)d_in[5];       // [1]
    float*           out = (float*)d_out;

    const int N = in_sizes[0] / 64;
    const int E = in_sizes[1] / 2;

    const size_t need = (size_t)N * 128 * sizeof(float);   // A[N,64] + B[N,64]
    if (ws_size < need) {
        edgemlp_fused_fallback<<<(E + 255) / 256, 256, 0, stream>>>(
            x, ei, W1, b1, W2, b2, out, E);
        return;
    }

    float* A = (float*)d_ws;
    float* B = A + (size_t)N * 64;

    const int ntiles = N / 16;
    if (ntiles > 0) {
        const int wavesPerBlock = 8;
        dim3 block(32 * wavesPerBlock);
        dim3 grid((ntiles + wavesPerBlock - 1) / wavesPerBlock);
        edgemlp_precompute_wmma<<<grid, block, 0, stream>>>(x, W1, A, B, ntiles);
    }
    const int rem = N & 15;
    if (rem) {
        edgemlp_precompute_tail<<<(rem * 64 + 63) / 64, 64, 0, stream>>>(
            x, W1, A, B, N - rem, rem);
    }

    edgemlp_edge_kernel<<<(E + 255) / 256, 256, 0, stream>>>(
        ei, A, B, b1, W2, b2, out, E);
}